// Multimodal_aggregator_59674275611127
// MI455X (gfx1250) — compile-verified
//
#include <hip/hip_runtime.h>
#include <hip/hip_bf16.h>

typedef __attribute__((ext_vector_type(2))) float v2f;
typedef __attribute__((ext_vector_type(8))) float v8f;

#define EMBED 512
#define NTOK  4108   // 4096 img + 8 table + 4 class
#define NIMG  4096
#define NKEY  4104   // img + table keys
#define NQ    12     // table + class queries
#define HD    64     // head dim
#define SCALE 0.125f // 64^-0.5

// d_out flat layout (reference tuple return order), in floats
#define OUT_OFF  0                        // out:          2*1*512      = 1024
#define AMOD_OFF 1024                     // A_mod:        2*8*4*4096   = 262144
#define AIMG_OFF (AMOD_OFF + 262144)      // A_img:        2*8*4096     = 65536
#define APRE_OFF (AIMG_OFF + 65536)       // A_pre:        2*8*4*4096   = 262144
#define AIT_OFF  (APRE_OFF + 262144)      // A_img_table:  2*8*4096     = 65536
#define ATC_OFF  (AIT_OFF + 65536)        // A_table_class:2*8*4        = 64

__device__ __forceinline__ float sigmoidf_(float x) {
    return 1.0f / (1.0f + __expf(-x));
}

// -------------------------------------------------------------------------
// Kernel 1: C[row, col] = sum_e X[b, rowBase+row, e] * W[wOff+col, e] + b[wOff+col]
// One wave per 16x64 tile: 4 accumulators, A reused across 4 N-subtiles.
// f32 WMMA 16x16x4, K = 512. Two-deep software pipeline: loads for k0+8 are
// issued before the WMMAs consuming k0, so VMEM latency overlaps the matrix
// pipe instead of a s_wait_loadcnt 0 in front of every WMMA.
// -------------------------------------------------------------------------
__global__ void gemm_qkv(const float* __restrict__ X, const float* __restrict__ W,
                         const float* __restrict__ Bv, float* __restrict__ Out,
                         int rowBase, int Mrows, int wOff, int outStrideB)
{
    const int lane   = threadIdx.x;
    const int lane15 = lane & 15;
    const int laneHi = lane >> 4;
    const int mt = blockIdx.x, nt = blockIdx.y, b = blockIdx.z;

    // A operand: lane holds row M = lane&15, VGPR j holds K = j + 2*laneHi
    const int rowA = rowBase + min(mt * 16 + lane15, Mrows - 1);
    const float* xrow = X + ((size_t)b * NTOK + rowA) * EMBED + 2 * laneHi;

    // B operand: lane holds col N = lane&15, VGPR j holds K = j + 2*laneHi
    const float* wp0 = W + (size_t)(wOff + nt * 64 +  0 + lane15) * EMBED + 2 * laneHi;
    const float* wp1 = W + (size_t)(wOff + nt * 64 + 16 + lane15) * EMBED + 2 * laneHi;
    const float* wp2 = W + (size_t)(wOff + nt * 64 + 32 + lane15) * EMBED + 2 * laneHi;
    const float* wp3 = W + (size_t)(wOff + nt * 64 + 48 + lane15) * EMBED + 2 * laneHi;

    // Init accumulators with bias (bias depends only on column = lane&15 per subtile)
    v8f acc0, acc1, acc2, acc3;
    {
        float b0 = Bv[wOff + nt * 64 +  0 + lane15];
        float b1 = Bv[wOff + nt * 64 + 16 + lane15];
        float b2 = Bv[wOff + nt * 64 + 32 + lane15];
        float b3 = Bv[wOff + nt * 64 + 48 + lane15];
#pragma unroll
        for (int r = 0; r < 8; ++r) { acc0[r] = b0; acc1[r] = b1; acc2[r] = b2; acc3[r] = b3; }
    }

    // Two pipeline stages in registers
    v2f aq[2];
    v2f bq[2][4];
    aq[0] = *(const v2f*)(xrow + 0);
    bq[0][0] = *(const v2f*)(wp0 + 0);
    bq[0][1] = *(const v2f*)(wp1 + 0);
    bq[0][2] = *(const v2f*)(wp2 + 0);
    bq[0][3] = *(const v2f*)(wp3 + 0);
    aq[1] = *(const v2f*)(xrow + 4);
    bq[1][0] = *(const v2f*)(wp0 + 4);
    bq[1][1] = *(const v2f*)(wp1 + 4);
    bq[1][2] = *(const v2f*)(wp2 + 4);
    bq[1][3] = *(const v2f*)(wp3 + 4);

    // Steady state: k0 = 0 .. EMBED-12, each iter loads k0+8 then consumes k0
#pragma unroll 2
    for (int k0 = 0; k0 < EMBED - 8; k0 += 4) {
        const int s = (k0 >> 2) & 1;
        const int kn = k0 + 8;
        v2f an  = *(const v2f*)(xrow + kn);
        v2f bn0 = *(const v2f*)(wp0 + kn);
        v2f bn1 = *(const v2f*)(wp1 + kn);
        v2f bn2 = *(const v2f*)(wp2 + kn);
        v2f bn3 = *(const v2f*)(wp3 + kn);
        acc0 = __builtin_amdgcn_wmma_f32_16x16x4_f32(false, aq[s], false, bq[s][0], (short)0, acc0, false, false);
        acc1 = __builtin_amdgcn_wmma_f32_16x16x4_f32(false, aq[s], false, bq[s][1], (short)0, acc1, false, false);
        acc2 = __builtin_amdgcn_wmma_f32_16x16x4_f32(false, aq[s], false, bq[s][2], (short)0, acc2, false, false);
        acc3 = __builtin_amdgcn_wmma_f32_16x16x4_f32(false, aq[s], false, bq[s][3], (short)0, acc3, false, false);
        aq[s] = an;
        bq[s][0] = bn0; bq[s][1] = bn1; bq[s][2] = bn2; bq[s][3] = bn3;
    }
    // Epilogue: remaining stages hold k0 = EMBED-8 (stage 0) and EMBED-4 (stage 1)
#pragma unroll
    for (int s = 0; s < 2; ++s) {
        acc0 = __builtin_amdgcn_wmma_f32_16x16x4_f32(false, aq[s], false, bq[s][0], (short)0, acc0, false, false);
        acc1 = __builtin_amdgcn_wmma_f32_16x16x4_f32(false, aq[s], false, bq[s][1], (short)0, acc1, false, false);
        acc2 = __builtin_amdgcn_wmma_f32_16x16x4_f32(false, aq[s], false, bq[s][2], (short)0, acc2, false, false);
        acc3 = __builtin_amdgcn_wmma_f32_16x16x4_f32(false, aq[s], false, bq[s][3], (short)0, acc3, false, false);
    }

    // C/D: lane holds col N = lane&15, VGPR r holds row M = r + 8*laneHi
#pragma unroll
    for (int r = 0; r < 8; ++r) {
        int row = mt * 16 + r + 8 * laneHi;
        if (row < Mrows) {
            float* orow = Out + (size_t)b * outStrideB + (size_t)row * EMBED + nt * 64 + lane15;
            orow[ 0] = acc0[r];
            orow[16] = acc1[r];
            orow[32] = acc2[r];
            orow[48] = acc3[r];
        }
    }
}

// -------------------------------------------------------------------------
// Kernel 2a: A_table_class[b,h,q] = mean over 8 table cols of
//            sigmoid(scale * Q[class q] . K[table tc]).  One wave per (b,h).
// -------------------------------------------------------------------------
__global__ void table_class(const float* __restrict__ Qb, const float* __restrict__ Kb,
                            float* __restrict__ Out)
{
    int bh = blockIdx.x, b = bh >> 3, h = bh & 7;
    int lane = threadIdx.x;
    int q = lane >> 3, tc = lane & 7;         // 4 class rows x 8 table cols
    const float* qp = Qb + ((size_t)b * NQ + 8 + q) * EMBED + h * HD;   // class queries = rows 8..11
    const float* kp = Kb + ((size_t)b * NKEY + NIMG + tc) * EMBED + h * HD;
    float dot = 0.f;
#pragma unroll 8
    for (int d = 0; d < HD; ++d) dot += qp[d] * kp[d];
    float s = sigmoidf_(SCALE * dot);
    // mean over the 8 table columns (groups of 8 lanes)
    s += __shfl_down(s, 4, 8);
    s += __shfl_down(s, 2, 8);
    s += __shfl_down(s, 1, 8);
    if (tc == 0) Out[ATC_OFF + bh * 4 + q] = s * 0.125f;
}

// -------------------------------------------------------------------------
// Kernel 2b: per 16-key image tile per (b,h): S = sigmoid(scale * Q_sub K^T)
// via 16 f32 WMMAs over d=64. All 32 operand fragments are loaded up front
// (one burst of 32 global_load_b64), then the WMMA chain runs with at most
// partial loadcnt waits. C/D layout puts the 8 table rows in lanes 0-15
// (column mean -> A_img_table) and the class rows in lanes 16-31
// (-> A_pre, A_mod, column max -> A_img).
// -------------------------------------------------------------------------
__global__ void img_scores(const float* __restrict__ Qb, const float* __restrict__ Kb,
                           float* __restrict__ Out)
{
    int kt = blockIdx.x, h = blockIdx.y, b = blockIdx.z;
    int bh = b * 8 + h;
    int lane = threadIdx.x, lane15 = lane & 15, laneHi = lane >> 4;
    int key0 = kt * 16;
    int c = key0 + lane15;

    const float* qrow = Qb + ((size_t)b * NQ + min(lane15, NQ - 1)) * EMBED + h * HD + 2 * laneHi;
    const float* krow = Kb + ((size_t)b * NKEY + c) * EMBED + h * HD + 2 * laneHi;

    // Prefetch the entire reduction: 16 A fragments + 16 B fragments
    v2f af[16], bf[16];
#pragma unroll
    for (int i = 0; i < 16; ++i) {
        af[i] = *(const v2f*)(qrow + 4 * i);
        bf[i] = *(const v2f*)(krow + 4 * i);
    }

    v8f acc = {};
#pragma unroll
    for (int i = 0; i < 16; ++i)
        acc = __builtin_amdgcn_wmma_f32_16x16x4_f32(false, af[i], false, bf[i], (short)0, acc, false, false);

    float s[8];
#pragma unroll
    for (int r = 0; r < 8; ++r) s[r] = sigmoidf_(SCALE * acc[r]);

    // Column mean over the 8 table rows (valid in lanes 0-15; garbage in high lanes)
    float it = (s[0] + s[1] + s[2] + s[3] + s[4] + s[5] + s[6] + s[7]) * 0.125f;
    // Broadcast the low-half value to both halves (ds_bpermute) BEFORE divergence
    float itb = __shfl(it, lane15);

    if (laneHi == 0) {
        Out[AIT_OFF + (size_t)bh * NIMG + c] = it;
    } else {
        // rows 8..11 = class queries q = 0..3 (vgprs 0..3); rows 12..15 discarded
        float amax = 0.f;
#pragma unroll
        for (int q = 0; q < 4; ++q) {
            float pre = s[q];
            Out[APRE_OFF + ((size_t)bh * 4 + q) * NIMG + c] = pre;
            float tcv = Out[ATC_OFF + bh * 4 + q];
            float am = pre * itb * tcv;
            Out[AMOD_OFF + ((size_t)bh * 4 + q) * NIMG + c] = am;
            amax = (q == 0) ? am : fmaxf(amax, am);
        }
        Out[AIMG_OFF + (size_t)bh * NIMG + c] = amax;
    }
}

// -------------------------------------------------------------------------
// Kernel 3: per (b,h): sum A_img, then out[d] = (1/(sum+1e-6)) * A_img . V[:,d]
// -------------------------------------------------------------------------
__global__ void finalize_out(const float* __restrict__ Vb, float* __restrict__ Out)
{
    __shared__ float red[256];
    int bh = blockIdx.x, b = bh >> 3, h = bh & 7;
    int tid = threadIdx.x;
    const float* aimg = Out + AIMG_OFF + (size_t)bh * NIMG;

    float part = 0.f;
    for (int c = tid; c < NIMG; c += 256) part += aimg[c];
    red[tid] = part;
    __syncthreads();
    for (int off = 128; off >= 1; off >>= 1) {
        if (tid < off) red[tid] += red[tid + off];
        __syncthreads();
    }
    float inv = 1.0f / (red[0] + 1e-6f);
    __syncthreads();

    int d = tid & 63, seg = tid >> 6;   // tid = seg*64 + d
    const float* vcol = Vb + (size_t)b * NIMG * EMBED + h * HD + d;
    float accu = 0.f;
#pragma unroll 4
    for (int j = 0; j < NIMG / 4; ++j) {
        int c = seg * (NIMG / 4) + j;
        accu += aimg[c] * vcol[(size_t)c * EMBED];
    }
    red[tid] = accu;
    __syncthreads();
    if (tid < 64) {
        float tot = red[tid] + red[tid + 64] + red[tid + 128] + red[tid + 192];
        Out[OUT_OFF + (size_t)b * EMBED + h * HD + tid] = tot * inv;
    }
}

// -------------------------------------------------------------------------
extern "C" void kernel_launch(void* const* d_in, const int* in_sizes, int n_in,
                              void* d_out, int out_size, void* d_ws, size_t ws_size,
                              hipStream_t stream) {
    const float* x  = (const float*)d_in[0];   // [2,4108,512]
    const float* w  = (const float*)d_in[1];   // [1536,512]
    const float* bq = (const float*)d_in[2];   // [1536]
    float* out = (float*)d_out;
    float* ws  = (float*)d_ws;

    float* Kb = ws;                                  // [2][4104][512]
    float* Vb = Kb + (size_t)2 * NKEY * EMBED;       // [2][4096][512]
    float* Qb = Vb + (size_t)2 * NIMG * EMBED;       // [2][12][512]

    // K projection: tokens [0,4104), weight rows [512,1024)
    gemm_qkv<<<dim3(257, 8, 2), 32, 0, stream>>>(x, w, bq, Kb, 0, NKEY, 512, NKEY * EMBED);
    // V projection: tokens [0,4096), weight rows [1024,1536)
    gemm_qkv<<<dim3(256, 8, 2), 32, 0, stream>>>(x, w, bq, Vb, 0, NIMG, 1024, NIMG * EMBED);
    // Q projection: tokens [4096,4108), weight rows [0,512)
    gemm_qkv<<<dim3(1, 8, 2), 32, 0, stream>>>(x, w, bq, Qb, NIMG, NQ, 0, NQ * EMBED);

    table_class<<<16, 32, 0, stream>>>(Qb, Kb, out);
    img_scores<<<dim3(256, 8, 2), 32, 0, stream>>>(Qb, Kb, out);
    finalize_out<<<16, 256, 0, stream>>>(Vb, out);
}